// CharSkipGramWaveModel_4440996184313
// MI455X (gfx1250) — compile-verified
//
#include <hip/hip_runtime.h>
#include <hip/hip_bf16.h>
#include <math.h>

// ---------------------------------------------------------------------------
// Problem constants (reference): V=128 chars, K=4 waves, B=1024 batch,
// 16 chars/word, 8 negatives. NF = V*K = 512 distinct (char,wave) components.
//
// Exact factorization: every pair energy is w_x^T S w_y where
//   w[c*4+k] = amp[c,k] * (#masked occurrences of char c in the word)
//   S[i][j]  = sinc(2*(F[i]-F[j]))   (512x512, computed once, L2-resident)
// Per batch element: Ut(16x512) = Wt(16x512) * S  via V_WMMA_F32_16X16X4_F32,
// then a 16x16 Gram G = Ut * Wt^T gives all 9 outputs.
// ---------------------------------------------------------------------------
#define VCAB   128
#define KWAV   4
#define NF     512
#define LW     16
#define NNEG   8
#define NWORDS 10
#define WPAD   516          // Wt row stride: conflict-free A gather (516%64==4)
#define SPAD   520          // S-block row stride: 2*520%64==16 -> halves disjoint
#define KB     16           // K-rows staged per async block (32 blocks total)

// LDS arena layout (floats)
#define WT_OFF   0
#define WT_SZ    (16 * WPAD)            // 8256
#define SB_OFF   (WT_OFF + WT_SZ)       // 8256
#define SB_SZ    (KB * SPAD)            // 8320 per buffer
#define UT_OFF   SB_OFF                 // Ut aliases S buffer 0 (post-GEMM only)
#define GS_OFF   (SB_OFF + 2 * SB_SZ)   // 24896
#define ARENA_SZ (GS_OFF + 256)         // 25152 floats = 100,608 B

typedef float v2f __attribute__((ext_vector_type(2)));
typedef float v8f __attribute__((ext_vector_type(8)));

// ---------------------------------------------------------------------------
// Kernel 1: S[i][j] = sinc(2*(F[i]-F[j])). 262144 transcendentals total,
// vs ~80M in the naive positional formulation.
// ---------------------------------------------------------------------------
__global__ void build_sinc_table(const float* __restrict__ freq,
                                 float* __restrict__ S) {
    int idx = blockIdx.x * blockDim.x + threadIdx.x;
    int i = idx >> 9;
    int j = idx & (NF - 1);
    float x  = 2.0f * (freq[i] - freq[j]);
    float px = 3.14159265358979323846f * x;
    S[idx] = (fabsf(px) < 1e-8f) ? 1.0f : (sinf(px) / px);
}

// ---------------------------------------------------------------------------
// Kernel 2: one workgroup (8 wave32) per batch element.
// Double-buffered GLOBAL_LOAD_ASYNC_TO_LDS_B128 streams 32KB S-blocks into
// LDS (ASYNCcnt-pipelined) while WMMA consumes the previous block.
// ---------------------------------------------------------------------------
__global__ __launch_bounds__(256)
void wave_energy_kernel(const int* __restrict__ tgt_c,
                        const int* __restrict__ pos_c,
                        const int* __restrict__ neg_c,
                        const int* __restrict__ tgt_m,
                        const int* __restrict__ pos_m,
                        const int* __restrict__ neg_m,
                        const float* __restrict__ amp,
                        const float* __restrict__ S,
                        float* __restrict__ out,
                        int B) {
    __shared__ float arena[ARENA_SZ];

    const int b    = blockIdx.x;
    const int t    = threadIdx.x;
    const int lane = t & 31;
    const int wave = t >> 5;
    const int l15  = lane & 15;             // A-row (M) / B-col (N) lane index
    const int kh   = (lane < 16) ? 0 : 2;   // K-half select (ISA A/B layout)
    const int xb   = (lane < 16) ? 0 : 8;   // D-row base (ISA C/D layout)

    float* Wt = &arena[WT_OFF];
    float* Gs = &arena[GS_OFF];

    // Per-thread async-copy geometry: 16 rows x 2048B, padded dst rows.
    // thread t owns row (t&15), column-chunks (t>>4) + 16*i, i=0..7.
    const int srow   = t & 15;
    const int schunk = t >> 4;
    const unsigned long long sbase = (unsigned long long)(uintptr_t)S;

    // ---- issue async staging of K-block 0 into buffer 0 ----
    {
        unsigned lds0 = (unsigned)(uintptr_t)&arena[SB_OFF + srow * SPAD];
        #pragma unroll
        for (int i = 0; i < 8; ++i) {
            unsigned goff = (unsigned)(srow * (NF * 4) + (schunk + 16 * i) * 16);
            unsigned loff = lds0 + (unsigned)((schunk + 16 * i) * 16);
            asm volatile("global_load_async_to_lds_b128 %0, %1, %2"
                         :: "v"(loff), "v"(goff), "s"(sbase) : "memory");
        }
    }

    // ---- zero Wt, then scatter-build the 10 word histograms ----
    for (int i = t; i < WT_SZ; i += 256) Wt[i] = 0.0f;
    __syncthreads();
    if (t < NWORDS * LW) {
        int w = t >> 4, p = t & 15;
        int id, m;
        if (w == 0)      { id = tgt_c[b * LW + p]; m = tgt_m[b * LW + p]; }
        else if (w == 1) { id = pos_c[b * LW + p]; m = pos_m[b * LW + p]; }
        else { int j = w - 2;
               id = neg_c[(b * NNEG + j) * LW + p];
               m  = neg_m[(b * NNEG + j) * LW + p]; }
        float mf = m ? 1.0f : 0.0f;
        #pragma unroll
        for (int k = 0; k < KWAV; ++k)
            atomicAdd(&Wt[w * WPAD + id * KWAV + k], amp[id * KWAV + k] * mf);
    }

    // ---- main pipeline: 32 K-blocks, double-buffered ----
    // wave owns N-range [wave*64, wave*64+64): 4 tiles of 16
    v8f acc0 = {}, acc1 = {}, acc2 = {}, acc3 = {};
    const int nbase = wave * 64;

    for (int blk = 0; blk < NF / KB; ++blk) {
        // issue next block's async loads into the other buffer
        if (blk + 1 < NF / KB) {
            unsigned ldsn = (unsigned)(uintptr_t)
                &arena[SB_OFF + ((blk + 1) & 1) * SB_SZ + srow * SPAD];
            unsigned gblk = (unsigned)((blk + 1) * KB * NF * 4);
            #pragma unroll
            for (int i = 0; i < 8; ++i) {
                unsigned goff = gblk + (unsigned)(srow * (NF * 4) +
                                                  (schunk + 16 * i) * 16);
                unsigned loff = ldsn + (unsigned)((schunk + 16 * i) * 16);
                asm volatile("global_load_async_to_lds_b128 %0, %1, %2"
                             :: "v"(loff), "v"(goff), "s"(sbase) : "memory");
            }
            // 8 newer ops in flight -> previous block's 8 have completed
            asm volatile("s_wait_asynccnt 0x8" ::: "memory");
        } else {
            asm volatile("s_wait_asynccnt 0x0" ::: "memory");
        }
        __syncthreads();   // whole staged block visible to all waves

        const float* Sb = &arena[SB_OFF + (blk & 1) * SB_SZ];
        #pragma unroll
        for (int kk = 0; kk < KB; kk += 4) {
            // A = Wt tile (16x4): lane->M=l15; VGPR0:K={0|2}, VGPR1:K={1|3}
            v2f a;
            const int kg = blk * KB + kk + kh;
            a.x = Wt[l15 * WPAD + kg];
            a.y = Wt[l15 * WPAD + kg + 1];
            // B = S tile (4x16) from LDS; one ds row per K, conflict-free pad
            v2f b0, b1, b2, b3;
            const int rb = (kk + kh) * SPAD + l15;
            b0.x = Sb[rb + nbase];        b0.y = Sb[rb + SPAD + nbase];
            b1.x = Sb[rb + nbase + 16];   b1.y = Sb[rb + SPAD + nbase + 16];
            b2.x = Sb[rb + nbase + 32];   b2.y = Sb[rb + SPAD + nbase + 32];
            b3.x = Sb[rb + nbase + 48];   b3.y = Sb[rb + SPAD + nbase + 48];
            acc0 = __builtin_amdgcn_wmma_f32_16x16x4_f32(false, a, false, b0,
                                                         (short)0, acc0, false, false);
            acc1 = __builtin_amdgcn_wmma_f32_16x16x4_f32(false, a, false, b1,
                                                         (short)0, acc1, false, false);
            acc2 = __builtin_amdgcn_wmma_f32_16x16x4_f32(false, a, false, b2,
                                                         (short)0, acc2, false, false);
            acc3 = __builtin_amdgcn_wmma_f32_16x16x4_f32(false, a, false, b3,
                                                         (short)0, acc3, false, false);
        }
        __syncthreads();   // block fully consumed; buffer may be overwritten
    }

    // ---- spill Ut (aliases S buffer 0; last block consumed buffer 1) ----
    float* Ut = &arena[UT_OFF];
    #pragma unroll
    for (int j = 0; j < 8; ++j) {
        Ut[(xb + j) * WPAD + nbase +  0 + l15] = acc0[j];
        Ut[(xb + j) * WPAD + nbase + 16 + l15] = acc1[j];
        Ut[(xb + j) * WPAD + nbase + 32 + l15] = acc2[j];
        Ut[(xb + j) * WPAD + nbase + 48 + l15] = acc3[j];
    }
    __syncthreads();

    // ---- Gram: G[x][y] = sum_i Ut[x][i] * Wt[y][i]  (= w_x^T S w_y) ----
    {
        const int x = t >> 4, y = t & 15;
        float g = 0.0f;
        for (int i = 0; i < NF; ++i)
            g = fmaf(Ut[x * WPAD + i], Wt[y * WPAD + i], g);
        Gs[t] = g;
    }
    __syncthreads();

    // ---- emit 9 outputs for this batch element ----
    if (t < 1 + NNEG) {
        float g00 = Gs[0];
        if (t == 0) {
            out[b] = g00 + Gs[1 * 16 + 1] + 2.0f * Gs[0 * 16 + 1];
        } else {
            int w = 1 + t;
            out[B + b * NNEG + (t - 1)] =
                g00 + Gs[w * 16 + w] + 2.0f * Gs[0 * 16 + w];
        }
    }
}

// ---------------------------------------------------------------------------
// Launch wrapper. d_in order: target_chars, pos_chars, neg_chars, target_mask,
// pos_mask, neg_mask, frequencies, amplitudes.
// d_out: [pos_energy (B)] ++ [neg_energy (B*8)], float32. d_ws: S (1 MB).
// ---------------------------------------------------------------------------
extern "C" void kernel_launch(void* const* d_in, const int* in_sizes, int n_in,
                              void* d_out, int out_size, void* d_ws, size_t ws_size,
                              hipStream_t stream) {
    const int*   tgt_c = (const int*)d_in[0];
    const int*   pos_c = (const int*)d_in[1];
    const int*   neg_c = (const int*)d_in[2];
    const int*   tgt_m = (const int*)d_in[3];
    const int*   pos_m = (const int*)d_in[4];
    const int*   neg_m = (const int*)d_in[5];
    const float* freq  = (const float*)d_in[6];
    const float* amp   = (const float*)d_in[7];

    const int B = in_sizes[0] / LW;
    float* S   = (float*)d_ws;
    float* out = (float*)d_out;

    build_sinc_table<<<(NF * NF) / 256, 256, 0, stream>>>(freq, S);
    wave_energy_kernel<<<B, 256, 0, stream>>>(tgt_c, pos_c, neg_c,
                                              tgt_m, pos_m, neg_m,
                                              amp, S, out, B);
}